// LogitsMaskLayer_34720515620877
// MI455X (gfx1250) — compile-verified
//
#include <hip/hip_runtime.h>
#include <stdint.h>

#define SEQ_LEN   32
#define BATCH     64
#define VOCAB     32000
#define SEP_TOKEN 2
// pattern {5,7,5,7,7,0} packed 3 bits each: (pat >> 3*seg) & 7
#define PAT_PACKED 32637u

typedef float        v4f   __attribute__((ext_vector_type(4)));
typedef unsigned int u32x4 __attribute__((ext_vector_type(4)));
typedef int          i32x4 __attribute__((ext_vector_type(4)));
typedef int          i32x8 __attribute__((ext_vector_type(8)));

// ---- TDM availability / arity detection (compile-time) ----
#if defined(__has_builtin)
#  if __has_builtin(__builtin_amdgcn_tensor_load_to_lds) && __has_builtin(__builtin_amdgcn_s_wait_tensorcnt)
#    define HAVE_TDM 1
#  endif
#endif
#ifndef HAVE_TDM
#  define HAVE_TDM 0
#endif
#if HAVE_TDM && defined(__has_include)
#  if __has_include(<hip/amd_detail/amd_gfx1250_TDM.h>)
#    define TDM_ARITY6 1
#  endif
#endif
#ifndef TDM_ARITY6
#  define TDM_ARITY6 0
#endif

// ------------------------------------------------------------------
// MAIN KERNEL FIRST IN FILE so the disasm snippet shows its prologue
// (need to verify tensor_load_to_lds actually lowered).
//
// Streaming mask: one block per (s,b) row of 32000 floats.
// u8 syllable table staged into LDS via TDM when available;
// logits/out use non-temporal B128 accesses (stream-once data).
// ------------------------------------------------------------------
__global__ __launch_bounds__(256)
void logits_mask_kernel(const float* __restrict__ logits,
                        const int* __restrict__ rs_buf,
                        const unsigned char* __restrict__ w2s_u8,
                        float* __restrict__ out) {
    __shared__ __align__(16) unsigned char s_tab[VOCAB];
    const int tid = threadIdx.x;
    const int row = blockIdx.x;               // row = s*BATCH + b

#if HAVE_TDM
    if (tid < 32) {                           // wave 0 issues one TDM descriptor
        unsigned long long ga = (unsigned long long)w2s_u8;
        unsigned lds_off = (unsigned)(unsigned long long)(const void*)s_tab;
        u32x4 g0;
        g0[0] = 1u;                                            // count=1, user mode
        g0[1] = lds_off;                                       // lds_addr
        g0[2] = (unsigned)(ga & 0xffffffffull);                // global_addr[31:0]
        g0[3] = (unsigned)((ga >> 32) & 0x1ffffffull) | (2u << 30); // ga[56:32], type=2
        i32x8 g1;
        g1[0] = 0;                               // wg_mask=0, data_size=0 (1 byte)
        g1[1] = (int)((unsigned)VOCAB << 16);    // tensor_dim0[15:0]
        g1[2] = (int)(1u << 16);                 // tensor_dim0 hi=0, tensor_dim1=1
        g1[3] = (int)((unsigned)VOCAB << 16);    // tensor_dim1 hi=0, tile_dim0=32000
        g1[4] = 1;                               // tile_dim1=1, tile_dim2=0
        g1[5] = VOCAB;                           // tensor_dim0_stride lo
        g1[6] = 0;
        g1[7] = 0;
        i32x4 gz = {0, 0, 0, 0};
#if TDM_ARITY6
        i32x8 gz8 = {0, 0, 0, 0, 0, 0, 0, 0};
        __builtin_amdgcn_tensor_load_to_lds(g0, g1, gz, gz, gz8, 0);
#else
        __builtin_amdgcn_tensor_load_to_lds(g0, g1, gz, gz, 0);
#endif
        __builtin_amdgcn_s_wait_tensorcnt(0);
    }
    __syncthreads();
#else
    {   // fallback: cooperative B128 copy of the 32 KB table into LDS
        const u32x4* g4 = (const u32x4*)w2s_u8;
        u32x4* s4 = (u32x4*)s_tab;
        for (int i = tid; i < VOCAB / 16; i += 256) s4[i] = g4[i];
        __syncthreads();
    }
#endif

    const int rs = rs_buf[row];                       // uniform -> scalar load
    const float* __restrict__ src = logits + (size_t)row * VOCAB;
    float*       __restrict__ dst = out    + (size_t)row * VOCAB;
    const unsigned int* __restrict__ tab32 = (const unsigned int*)s_tab;
    const float NEG_INF = -__builtin_inff();

#pragma unroll 4
    for (int i = tid; i < VOCAB / 4; i += 256) {
        v4f v = __builtin_nontemporal_load((const v4f*)src + i);  // global_load_b128 NT
        unsigned int p = tab32[i];                                // ds_load_b32: 4 syllable bytes
        v.x = ((int)( p        & 0xff) > rs) ? NEG_INF : v.x;
        v.y = ((int)((p >>  8) & 0xff) > rs) ? NEG_INF : v.y;
        v.z = ((int)((p >> 16) & 0xff) > rs) ? NEG_INF : v.z;
        v.w = ((int)( p >> 24        ) > rs) ? NEG_INF : v.w;
        __builtin_nontemporal_store(v, (v4f*)dst + i);            // global_store_b128 NT
    }
}

// ------------------------------------------------------------------
// Sequential haiku-syllable scan, one lane per batch column.
// rs_out: [S, B] int32 in workspace.
// ------------------------------------------------------------------
__global__ void rs_scan_kernel(const int* __restrict__ dec,
                               const int* __restrict__ w2s,
                               int* __restrict__ rs_out) {
    int b = threadIdx.x;
    if (b >= BATCH) return;
    int inp0 = dec[b];
    int seg  = (inp0 == SEP_TOKEN) ? 1 : 0;
    int rs   = (int)((PAT_PACKED >> (3 * seg)) & 7u);
    rs_out[b] = rs;
    for (int s = 1; s < SEQ_LEN; ++s) {
        int inp = dec[s * BATCH + b];
        rs -= w2s[inp];
        if (rs < 0) rs = 0;
        if (inp == SEP_TOKEN) {
            seg = (seg >= 5) ? 5 : seg + 1;
            rs  = (int)((PAT_PACKED >> (3 * seg)) & 7u);
        }
        rs_out[s * BATCH + b] = rs;
    }
}

// ------------------------------------------------------------------
// Compact word2syllables (values 0..4) int32 -> u8 table.
// ------------------------------------------------------------------
__global__ void w2s_compact_kernel(const int* __restrict__ w2s,
                                   unsigned char* __restrict__ out_u8) {
    int i = blockIdx.x * blockDim.x + threadIdx.x;
    if (i < VOCAB) out_u8[i] = (unsigned char)w2s[i];
}

// ------------------------------------------------------------------
// Launch: inputs are {logits, decoder_input, word2syllables, n}.
// d_ws layout: [0, 8192)   rs int32[S*B]
//              [8192, ...) u8 syllable table[VOCAB]
// ------------------------------------------------------------------
extern "C" void kernel_launch(void* const* d_in, const int* in_sizes, int n_in,
                              void* d_out, int out_size, void* d_ws, size_t ws_size,
                              hipStream_t stream) {
    (void)in_sizes; (void)n_in; (void)out_size; (void)ws_size;
    const float* logits = (const float*)d_in[0];
    const int*   dec    = (const int*)d_in[1];
    const int*   w2s    = (const int*)d_in[2];
    float*       out    = (float*)d_out;

    int*           rs_buf = (int*)d_ws;
    unsigned char* w2s_u8 = (unsigned char*)d_ws + 8192;

    rs_scan_kernel<<<1, 64, 0, stream>>>(dec, w2s, rs_buf);
    w2s_compact_kernel<<<(VOCAB + 255) / 256, 256, 0, stream>>>(w2s, w2s_u8);
    logits_mask_kernel<<<SEQ_LEN * BATCH, 256, 0, stream>>>(logits, rs_buf, w2s_u8, out);
}